// SelectNet_2731599200721
// MI455X (gfx1250) — compile-verified
//
#include <hip/hip_runtime.h>
#include <hip/hip_bf16.h>
#include <stdint.h>

#define CROP   81
#define WIN    86          // staged window (covers 82-tap span + floor-rounding slack)
#define NIMG   64
#define NP     6
#define HH     512
#define WW     512
#define PLANE  (WIN*WIN)                 // elements per staged plane
#define SMEM_BYTES (4*PLANE*4)           // 3 img channels + 1 label plane, 4B each

typedef unsigned int u32x4 __attribute__((ext_vector_type(4)));
typedef int          i32x4 __attribute__((ext_vector_type(4)));
typedef int          i32x8 __attribute__((ext_vector_type(8)));

static __device__ __forceinline__ unsigned rfl(unsigned x) {
  return (unsigned)__builtin_amdgcn_readfirstlane((int)x);
}
static __device__ __forceinline__ int iclamp(int v, int lo, int hi) {
  return v < lo ? lo : (v > hi ? hi : v);
}

// Issue one TDM 2D-tile load: WIN x WIN tile of 4-byte elements,
// row stride = 512 elements, from global byte address gaddr into LDS offset lds_addr.
// D# bitfield packing per CDNA5 ISA 8.3-8.6.  6-arg builtin form (clang-23 /
// therock-10.0 headers): (v4u g0, v8i g1, v4i g2, v4i g3, v8i extra, i32 cpol).
static __device__ __forceinline__ void tdm_load_2d(unsigned lds_addr, uint64_t gaddr) {
  u32x4 g0;
  g0[0] = 1u;                                                    // count=1, user descriptor
  g0[1] = rfl(lds_addr);                                         // lds_addr (bytes)
  g0[2] = rfl((unsigned)(gaddr & 0xFFFFFFFFull));                // global_addr[31:0]
  g0[3] = rfl((unsigned)((gaddr >> 32) & 0x1FFFFFFull)) | (2u << 30); // addr[56:32] | type=2

  i32x8 g1;
  g1[0] = (int)(2u << 16);                 // workgroup_mask=0, data_size=4B, no iterate/pad
  g1[1] = 0;                               // barrier addr=0, tensor_dim0[15:0]=0 (dim0=2^20)
  g1[2] = 0x10;                            // tensor_dim0[31:16]=0x10, tensor_dim1[15:0]=0
  g1[3] = (int)(0x10u | ((unsigned)WIN << 16)); // tensor_dim1[31:16]=0x10, tile_dim0=WIN
  g1[4] = WIN;                             // tile_dim1=WIN, tile_dim2=0 (2D tile)
  g1[5] = WW;                              // tensor_dim0_stride[31:0] = 512 elements
  g1[6] = 0;                               // stride0[47:32]=0, stride1[15:0]=0
  g1[7] = 4;                               // stride1[47:16] (stride1 = 262144 elements)

  i32x4 g2; g2[0] = (int)(1u << 20); g2[1] = 0; g2[2] = 0; g2[3] = 0; // tile_dim3=0 -> unused
  i32x4 g3; g3[0] = 0; g3[1] = 0; g3[2] = 0; g3[3] = 0;
  i32x8 gx; gx[0]=0; gx[1]=0; gx[2]=0; gx[3]=0; gx[4]=0; gx[5]=0; gx[6]=0; gx[7]=0;

  __builtin_amdgcn_tensor_load_to_lds(g0, g1, g2, g3, gx, 0);
}

extern "C" __global__ void __launch_bounds__(256)
selectnet_kernel(const float* __restrict__ img,
                 const int*   __restrict__ label,
                 const float* __restrict__ pts,
                 float*       __restrict__ out)
{
  extern __shared__ char smemRaw[];
  float* sImg = (float*)smemRaw;
  int*   sLab = (int*)(smemRaw + 3*PLANE*4);

  const int blk = blockIdx.x;
  const int n   = blk / NP;
  const int p   = blk - n*NP;
  const int tid = threadIdx.x;

  // crop-center point (reference: pts6 = [points[1:6], mean(points[6:9])])
  float py, px;
  if (p < 5) {
    const float* q = pts + ((size_t)n*9 + (1 + p))*2;
    py = q[0]; px = q[1];
  } else {
    const float* q = pts + ((size_t)n*9 + 6)*2;
    py = (q[0] + q[2] + q[4]) * (1.0f/3.0f);
    px = (q[1] + q[3] + q[5]) * (1.0f/3.0f);
  }

  const float sx = (float)(CROP - 1) / (float)(WW - 1);   // 80/511
  const float sy = (float)CROP / (float)HH;               // 81/512
  const float tx = -1.0f + 2.0f*px/(float)(WW - 1);
  const float ty = -1.0f + 2.0f*py/(float)(HH - 1);
  const float Kx = (tx + 1.0f)*0.5f*(float)(WW - 1);      // ix at base=0
  const float Ky = (ty + 1.0f)*0.5f*(float)(HH - 1);

  // window start; sample span is <=84 wide, WIN=86 leaves slack; clamp fully in-image
  int xs = iclamp((int)floorf(Kx) - 42, 0, WW - WIN);
  int ys = iclamp((int)floorf(Ky) - 42, 0, HH - WIN);

  // --- stage window via Tensor Data Mover (async-tensor path) ---
  if (tid == 0) {
    unsigned ldsBase = (unsigned)(size_t)smemRaw;
    uint64_t imgBase = (uint64_t)(uintptr_t)img;
    #pragma unroll
    for (int c = 0; c < 3; ++c) {
      uint64_t ga = imgBase + 4ull*((((uint64_t)n*3 + (uint64_t)c)*HH + (uint64_t)ys)*WW + (uint64_t)xs);
      tdm_load_2d(ldsBase + (unsigned)(c*PLANE*4), ga);
    }
    uint64_t la = (uint64_t)(uintptr_t)label + 4ull*(((uint64_t)n*HH + (uint64_t)ys)*WW + (uint64_t)xs);
    tdm_load_2d(ldsBase + (unsigned)(3*PLANE*4), la);
    __builtin_amdgcn_s_wait_tensorcnt(0);
  }
  __syncthreads();

  float* outImg  = out;
  float* outPred = out + (size_t)NIMG*NP*3*CROP*CROP;
  const size_t imgB  = (size_t)(n*NP + p)*3*(CROP*CROP);
  const size_t predB = (size_t)(n*NP + p)*(CROP*CROP);
  const float  clsv  = (float)(p + 1);

  for (int pix = tid; pix < CROP*CROP; pix += 256) {
    const int i = pix / CROP;
    const int j = pix - i*CROP;
    const float bj = -1.0f + 0.025f*(float)j;
    const float bi = -1.0f + 0.025f*(float)i;
    const float gx = sx*bj + tx;
    const float gy = sy*bi + ty;
    const float ix = (gx + 1.0f)*0.5f*(float)(WW - 1);
    const float iy = (gy + 1.0f)*0.5f*(float)(HH - 1);

    // ---- bilinear image sample from LDS ----
    const float x0f = floorf(ix), y0f = floorf(iy);
    const float fx = ix - x0f,  fy = iy - y0f;
    const int x0 = (int)x0f, y0 = (int)y0f;
    const float wgt00 = (1.0f - fx)*(1.0f - fy);
    const float wgt10 = fx*(1.0f - fy);
    const float wgt01 = (1.0f - fx)*fy;
    const float wgt11 = fx*fy;
    float a0 = 0.0f, a1 = 0.0f, a2 = 0.0f;
    #pragma unroll
    for (int dy = 0; dy < 2; ++dy) {
      const int  yc = y0 + dy;
      const bool vy = (yc >= 0) && (yc <= HH - 1);
      const int  yi = iclamp(yc, 0, HH - 1);
      const int  rb = (yi - ys)*WIN;
      #pragma unroll
      for (int dx = 0; dx < 2; ++dx) {
        const int  xc = x0 + dx;
        const bool v  = vy && (xc >= 0) && (xc <= WW - 1);
        const int  xi = iclamp(xc, 0, WW - 1);
        const int  off = rb + (xi - xs);
        const float w = v ? (dy ? (dx ? wgt11 : wgt01) : (dx ? wgt10 : wgt00)) : 0.0f;
        a0 += w * sImg[off];
        a1 += w * sImg[PLANE + off];
        a2 += w * sImg[2*PLANE + off];
      }
    }
    outImg[imgB + pix]               = a0;
    outImg[imgB + CROP*CROP + pix]   = a1;
    outImg[imgB + 2*CROP*CROP + pix] = a2;

    // ---- nearest label sample (round-half-even, matches jnp.round) ----
    const float xr = rintf(ix), yr = rintf(iy);
    const bool lv = (xr >= 0.0f) && (xr <= (float)(WW - 1)) &&
                    (yr >= 0.0f) && (yr <= (float)(HH - 1));
    const int xl = iclamp((int)xr, 0, WW - 1);
    const int yl = iclamp((int)yr, 0, HH - 1);
    const float labf = lv ? (float)sLab[(yl - ys)*WIN + (xl - xs)] : 0.0f;

    float pred;
    if (p < 5) {
      pred = (labf == clsv) ? 1.0f : 0.0f;
    } else {
      pred = (labf == 6.0f) ? 1.0f :
             (labf == 7.0f) ? 2.0f :
             (labf == 8.0f) ? 3.0f : 0.0f;
    }
    outPred[predB + pix] = pred;
  }
}

extern "C" void kernel_launch(void* const* d_in, const int* in_sizes, int n_in,
                              void* d_out, int out_size, void* d_ws, size_t ws_size,
                              hipStream_t stream) {
  (void)in_sizes; (void)n_in; (void)out_size; (void)d_ws; (void)ws_size;
  const float* img   = (const float*)d_in[0];
  const int*   label = (const int*)d_in[1];
  const float* pts   = (const float*)d_in[2];
  float*       out   = (float*)d_out;

  dim3 grid(NIMG*NP);   // one workgroup per (image, crop)
  dim3 block(256);      // 8 wave32
  selectnet_kernel<<<grid, block, SMEM_BYTES, stream>>>(img, label, pts, out);
}